// mk_lv1_84859963834782
// MI455X (gfx1250) — compile-verified
//
#include <hip/hip_runtime.h>

typedef __attribute__((ext_vector_type(2))) float v2f;
typedef __attribute__((ext_vector_type(8))) float v8f;

#define NGI  13
#define NLOC 10

// One wave handles 16 batch elements ("tile"). Per lane: m = lane&15 selects the
// batch row, khalf = lane>>4 selects which K-pair (g mod 4 = 0/1 or 2/3) the lane
// accumulates, matching the f32 WMMA A-matrix layout (16x4: lanes 0-15 -> K0/K1,
// lanes 16-31 -> K2/K3). A single v_wmma_f32_16x16x4_f32 against an all-ones B
// then sums the 16 padded g-slots per batch row in f32.
__global__ __launch_bounds__(256)
void diagrar_wmma_kernel(const float* __restrict__ e_i,
                         const float* __restrict__ r1_1,
                         const float* __restrict__ kv,
                         const float* __restrict__ dt,
                         const float* __restrict__ n,
                         const float* __restrict__ nx,
                         const float* __restrict__ detwei,
                         const float* __restrict__ Rv,
                         float* __restrict__ out,
                         int nB)
{
    __shared__ float s_nr2[16];   // (n[g,:]·R)^2, zero-padded to 16

    const int tid = threadIdx.x;
    if (tid < 16) {
        float val = 0.0f;
        if (tid < NGI) {
            float acc = 0.0f;
            #pragma unroll
            for (int i = 0; i < NLOC; ++i) acc = fmaf(n[tid * NLOC + i], Rv[i], acc);
            val = acc * acc;
        }
        s_nr2[tid] = val;
    }
    __syncthreads();

    const int lane = tid & 31;
    const int wave = tid >> 5;
    const int tile = blockIdx.x * 8 + wave;   // 8 waves / 256-thread block
    const int b0   = tile * 16;
    if (b0 >= nB) return;                     // wave-uniform exit: EXEC stays all-1s

    const float inv_dt = 1.0f / dt[0];

    float Rr[NLOC];
    #pragma unroll
    for (int i = 0; i < NLOC; ++i) Rr[i] = Rv[i];   // uniform -> scalar loads

    const int   m     = lane & 15;
    const int   khalf = lane >> 4;
    const int   b     = b0 + m;
    const float vmask = (b < nB) ? 1.0f : 0.0f;
    const int   bc    = (b < nB) ? b : (nB - 1);    // clamp to keep loads in-bounds
    const float kb    = kv[bc];

    const float* nx_b = nx + (size_t)bc * (2 * NGI * NLOC);

    float a0 = 0.0f, a1 = 0.0f;   // accumulators for this lane's two K slots
    #pragma unroll
    for (int c4 = 0; c4 < 4; ++c4) {
        #pragma unroll
        for (int j = 0; j < 2; ++j) {
            const int g = c4 * 4 + khalf * 2 + j;
            if (g < NGI) {
                const float* row0 = nx_b + g * NLOC;      // nx[b,0,g,:]
                const float* row1 = row0 + NGI * NLOC;    // nx[b,1,g,:]
                if (g + 4 < NGI) {                        // prefetch next chunk's rows
                    __builtin_prefetch(row0 + 4 * NLOC, 0, 1);
                    __builtin_prefetch(row1 + 4 * NLOC, 0, 1);
                }
                float d1 = 0.0f, d2 = 0.0f;
                #pragma unroll
                for (int i = 0; i < NLOC / 2; ++i) {      // rows are 8B aligned
                    v2f p0 = __builtin_nontemporal_load((const v2f*)row0 + i);
                    v2f p1 = __builtin_nontemporal_load((const v2f*)row1 + i);
                    d1 = fmaf(p0[0], Rr[2 * i], d1);
                    d1 = fmaf(p0[1], Rr[2 * i + 1], d1);
                    d2 = fmaf(p1[0], Rr[2 * i], d2);
                    d2 = fmaf(p1[1], Rr[2 * i + 1], d2);
                }
                const float dw = __builtin_nontemporal_load(detwei + (size_t)bc * NGI + g);
                const float q  = d1 * d1 + d2 * d2;
                const float z  = vmask * dw * fmaf(kb, q, s_nr2[g] * inv_dt);
                if (j == 0) a0 += z; else a1 += z;
            }
        }
    }

    // D = A x ones(4x16): every column of D equals the per-row K-sum -> diagRAR.
    v2f A;  A[0] = a0;   A[1] = a1;
    v2f Bm; Bm[0] = 1.0f; Bm[1] = 1.0f;
    v8f C = {};
    C = __builtin_amdgcn_wmma_f32_16x16x4_f32(
            /*neg_a=*/false, A, /*neg_b=*/false, Bm,
            /*c_mod=*/(short)0, C, /*reuse_a=*/false, /*reuse_b=*/false);

    // C/D layout: lanes 0-15 vgpr r -> M=r ; lanes 16-31 vgpr r -> M=8+r.
    // All columns equal, so lanes 0-7 and 16-23 each own one distinct M.
    const int r = lane & 7;
    float s0 = (r & 1) ? C[1] : C[0];
    float s1 = (r & 1) ? C[3] : C[2];
    float s2 = (r & 1) ? C[5] : C[4];
    float s3 = (r & 1) ? C[7] : C[6];
    float t0 = (r & 2) ? s1 : s0;
    float t1 = (r & 2) ? s3 : s2;
    float diag = (r & 4) ? t1 : t0;

    const int mm = r + khalf * 8;
    const int bo = b0 + mm;
    if (((lane & 8) == 0) && bo < nB) {
        out[bo] = diag;                                    // diagRAR
        const float ev = __builtin_nontemporal_load(e_i  + bo);
        const float rv = __builtin_nontemporal_load(r1_1 + bo);
        out[(size_t)nB + bo] = rv - diag * ev;             // rr1
    }
}

extern "C" void kernel_launch(void* const* d_in, const int* in_sizes, int n_in,
                              void* d_out, int out_size, void* d_ws, size_t ws_size,
                              hipStream_t stream)
{
    const float* e_i    = (const float*)d_in[0];
    const float* r1_1   = (const float*)d_in[1];
    const float* k      = (const float*)d_in[2];
    const float* dt     = (const float*)d_in[3];
    const float* n      = (const float*)d_in[4];
    const float* nx     = (const float*)d_in[5];
    const float* detwei = (const float*)d_in[6];
    const float* R      = (const float*)d_in[7];
    float* out = (float*)d_out;

    const int nB     = in_sizes[0];          // B = 500000
    const int tiles  = (nB + 15) / 16;       // 16 batch elems per wave
    const int blocks = (tiles + 7) / 8;      // 8 waves per 256-thread block

    hipLaunchKernelGGL(diagrar_wmma_kernel, dim3(blocks), dim3(256), 0, stream,
                       e_i, r1_1, k, dt, n, nx, detwei, R, out, nB);
}